// Attention_12747462934680
// MI455X (gfx1250) — compile-verified
//
#include <hip/hip_runtime.h>
#include <math.h>

#define Bc 64
#define Lc 2048
#define Hc 512

typedef __attribute__((ext_vector_type(2))) float v2f;
typedef __attribute__((ext_vector_type(8))) float v8f;

// Native gfx1250 V_TANH_F32 if the toolchain exposes it; safe fallback otherwise.
#if defined(__has_builtin)
#  if __has_builtin(__builtin_amdgcn_tanhf)
#    define FAST_TANH(x) __builtin_amdgcn_tanhf(x)
#  else
#    define FAST_TANH(x) tanhf(x)
#  endif
#else
#  define FAST_TANH(x) tanhf(x)
#endif

#define WMMA_F32(A, B, C) \
  __builtin_amdgcn_wmma_f32_16x16x4_f32(false, (A), false, (B), (short)0, (C), false, false)

// ---------------------------------------------------------------------------
// Kernel 1: hp[b,h] = bias[h] + hidden[b,:] . W[h, 0:H]   (rank-B half of GEMM)
// ---------------------------------------------------------------------------
__global__ __launch_bounds__(256)
void hidden_proj(const float* __restrict__ hidden,
                 const float* __restrict__ W,
                 const float* __restrict__ bias,
                 float* __restrict__ hp) {
  __shared__ float sh[Hc];
  const int b = blockIdx.x;
  for (int i = threadIdx.x; i < Hc; i += 256) sh[i] = hidden[(size_t)b * Hc + i];
  __syncthreads();
  for (int h = threadIdx.x; h < Hc; h += 256) {
    const float* wr = W + (size_t)h * (2 * Hc);   // first H columns = W_h row
    float acc = bias[h];
#pragma unroll 8
    for (int k = 0; k < Hc; ++k) acc = fmaf(sh[k], wr[k], acc);
    hp[(size_t)b * Hc + h] = acc;
  }
}

// ---------------------------------------------------------------------------
// Kernel 2: main f32 WMMA GEMM (enc @ W_e^T), fused tanh + v-weighted reduce.
// Workgroup: 8 waves, 32-row M tile (2 row-tiles/wave), full N=512
// (each wave owns 64 columns -> 8 accumulator tiles, 6 loads per 8 WMMAs).
// ---------------------------------------------------------------------------
__global__ __launch_bounds__(256)
void attn_gemm_score(const float* __restrict__ enc,
                     const float* __restrict__ W,
                     const float* __restrict__ v,
                     const float* __restrict__ hp,
                     float* __restrict__ scores) {
  __shared__ float s_score[32];

  const int tid    = threadIdx.x;
  const int wave   = tid >> 5;
  const int lane   = tid & 31;
  const int lane15 = lane & 15;
  const int khalf  = (lane >> 4) << 1;        // 0 or 2: K sub-pair this half holds
  const int rowlo  = (lane >> 4) << 3;        // C rows: g (lanes 0-15) / g+8 (16-31)

  const size_t m0 = (size_t)blockIdx.x * 32;  // global row = b*L + l  (32 | 2048)
  const int b  = (int)(m0 / Lc);
  const int l0 = (int)(m0 % Lc);

  if (tid < 32) s_score[tid] = 0.0f;
  __syncthreads();

  // A fragment sources: rows shared by all 8 waves -> WGP$ hits
  const float* Arow0 = enc + (m0 + (size_t)lane15) * Hc;        // rows m0..m0+15
  const float* Arow1 = Arow0 + (size_t)16 * Hc;                 // rows m0+16..m0+31

  // B fragment source: W_e rows (n = output h index), contiguous along k
  const int n_base = wave * 64;
  const size_t Wst = 2 * Hc;
  const float* Bb  = W + Hc;                  // encoder half of W
  const float* Brow0 = Bb + (size_t)(n_base +  0 + lane15) * Wst;
  const float* Brow1 = Bb + (size_t)(n_base + 16 + lane15) * Wst;
  const float* Brow2 = Bb + (size_t)(n_base + 32 + lane15) * Wst;
  const float* Brow3 = Bb + (size_t)(n_base + 48 + lane15) * Wst;

  v8f a00 = {}, a01 = {}, a02 = {}, a03 = {};   // m-tile 0, n-tiles 0..3
  v8f a10 = {}, a11 = {}, a12 = {}, a13 = {};   // m-tile 1, n-tiles 0..3

#pragma unroll 2
  for (int k = 0; k < Hc; k += 4) {
    v2f fa0 = *(const v2f*)(Arow0 + k + khalf);
    v2f fa1 = *(const v2f*)(Arow1 + k + khalf);
    v2f fb0 = *(const v2f*)(Brow0 + k + khalf);
    v2f fb1 = *(const v2f*)(Brow1 + k + khalf);
    v2f fb2 = *(const v2f*)(Brow2 + k + khalf);
    v2f fb3 = *(const v2f*)(Brow3 + k + khalf);
    a00 = WMMA_F32(fa0, fb0, a00);
    a10 = WMMA_F32(fa1, fb0, a10);
    a01 = WMMA_F32(fa0, fb1, a01);
    a11 = WMMA_F32(fa1, fb1, a11);
    a02 = WMMA_F32(fa0, fb2, a02);
    a12 = WMMA_F32(fa1, fb2, a12);
    a03 = WMMA_F32(fa0, fb3, a03);
    a13 = WMMA_F32(fa1, fb3, a13);
  }

  // Fused epilogue: contrib[g] = sum_j v[n_j] * tanh(acc_j[g] + hp[b,n_j])
  float c0[8], c1[8];
#pragma unroll
  for (int g = 0; g < 8; ++g) { c0[g] = 0.0f; c1[g] = 0.0f; }

  const v8f* t0[4] = { &a00, &a01, &a02, &a03 };
  const v8f* t1[4] = { &a10, &a11, &a12, &a13 };
#pragma unroll
  for (int j = 0; j < 4; ++j) {
    const int n   = n_base + j * 16 + lane15;
    const float h = hp[(size_t)b * Hc + n];
    const float w = v[n];
    const v8f  x0 = *t0[j];
    const v8f  x1 = *t1[j];
#pragma unroll
    for (int g = 0; g < 8; ++g) {
      c0[g] = fmaf(w, FAST_TANH(x0[g] + h), c0[g]);
      c1[g] = fmaf(w, FAST_TANH(x1[g] + h), c1[g]);
    }
  }

  // Reduce the 16 columns held by each half-wave, then LDS-atomic into rows.
#pragma unroll
  for (int g = 0; g < 8; ++g) {
    float s0 = c0[g];
    float s1 = c1[g];
    s0 += __shfl_xor(s0, 1, 32);  s1 += __shfl_xor(s1, 1, 32);
    s0 += __shfl_xor(s0, 2, 32);  s1 += __shfl_xor(s1, 2, 32);
    s0 += __shfl_xor(s0, 4, 32);  s1 += __shfl_xor(s1, 4, 32);
    s0 += __shfl_xor(s0, 8, 32);  s1 += __shfl_xor(s1, 8, 32);
    if (lane15 == 0) {
      atomicAdd(&s_score[g + rowlo], s0);        // ds_add_f32
      atomicAdd(&s_score[16 + g + rowlo], s1);
    }
  }
  __syncthreads();

  if (tid < 32) scores[(size_t)b * Lc + l0 + tid] = s_score[tid];
}

// ---------------------------------------------------------------------------
// Kernel 3: softmax over L per batch row -> out [B,1,L]
// ---------------------------------------------------------------------------
__global__ __launch_bounds__(256)
void softmax_rows(const float* __restrict__ scores, float* __restrict__ out) {
  __shared__ float red[256];
  const int b = blockIdx.x;
  const float* s = scores + (size_t)b * Lc;

  float m = -INFINITY;
  for (int i = threadIdx.x; i < Lc; i += 256) m = fmaxf(m, s[i]);
  red[threadIdx.x] = m;
  __syncthreads();
  for (int off = 128; off > 0; off >>= 1) {
    if (threadIdx.x < off) red[threadIdx.x] = fmaxf(red[threadIdx.x], red[threadIdx.x + off]);
    __syncthreads();
  }
  m = red[0];
  __syncthreads();

  float sum = 0.0f;
  for (int i = threadIdx.x; i < Lc; i += 256) sum += __expf(s[i] - m);
  red[threadIdx.x] = sum;
  __syncthreads();
  for (int off = 128; off > 0; off >>= 1) {
    if (threadIdx.x < off) red[threadIdx.x] += red[threadIdx.x + off];
    __syncthreads();
  }
  const float inv = 1.0f / red[0];

  for (int i = threadIdx.x; i < Lc; i += 256)
    out[(size_t)b * Lc + i] = __expf(s[i] - m) * inv;
}

// ---------------------------------------------------------------------------
extern "C" void kernel_launch(void* const* d_in, const int* in_sizes, int n_in,
                              void* d_out, int out_size, void* d_ws, size_t ws_size,
                              hipStream_t stream) {
  (void)in_sizes; (void)n_in; (void)out_size; (void)ws_size;

  const float* hidden = (const float*)d_in[0];   // [B, H]
  const float* enc    = (const float*)d_in[1];   // [B, L, H]
  const float* W      = (const float*)d_in[2];   // [H, 2H]
  const float* bias   = (const float*)d_in[3];   // [H]
  const float* v      = (const float*)d_in[4];   // [H]
  float*       out    = (float*)d_out;           // [B, 1, L]

  float* hp     = (float*)d_ws;                  // [B, H]   = 128 KB
  float* scores = hp + (size_t)Bc * Hc;          // [B, L]   = 512 KB

  hidden_proj    <<<Bc,              256, 0, stream>>>(hidden, W, bias, hp);
  attn_gemm_score<<<(Bc * Lc) / 32,  256, 0, stream>>>(enc, W, v, hp, scores);
  softmax_rows   <<<Bc,              256, 0, stream>>>(scores, out);
}